// MaximalCodingRateReduction_16406775070860
// MI455X (gfx1250) — compile-verified
//
#include <hip/hip_runtime.h>
#include <hip/hip_bf16.h>
#include <math.h>

#define K_CLASSES 32
#define D_FEAT    128
#define GAM1f     1.0f
#define GAM2f     1.0f
#define EPSf      0.5f

typedef float v2f __attribute__((ext_vector_type(2)));
typedef float v8f __attribute__((ext_vector_type(8)));
typedef int   v4i __attribute__((ext_vector_type(4)));

#define AS1 __attribute__((address_space(1)))
#define AS3 __attribute__((address_space(3)))

#if defined(__has_builtin)
#  if __has_builtin(__builtin_amdgcn_global_load_async_to_lds_b128)
#    define HAVE_ASYNC_LDS 1
#  endif
#endif
#ifndef HAVE_ASYNC_LDS
#  define HAVE_ASYNC_LDS 0
#endif

__device__ __forceinline__ void wait_async0() {
#if HAVE_ASYNC_LDS
#  if __has_builtin(__builtin_amdgcn_s_wait_asynccnt)
    __builtin_amdgcn_s_wait_asynccnt(0);
#  else
    asm volatile("s_wait_asynccnt 0x0" ::: "memory");
#  endif
#endif
}

// ---------------- kernel 1: zero output + ws control block -----------------
__global__ void k_zero(float* __restrict__ out, int out_n, float* __restrict__ wsf) {
    int i = blockIdx.x * blockDim.x + threadIdx.x;
    int stride = gridDim.x * blockDim.x;
    for (int j = i; j < out_n; j += stride) out[j] = 0.f;
    if (i < 512) wsf[i] = 0.f;
}

// ---------------- kernel 2: class histogram --------------------------------
__global__ __launch_bounds__(256) void k_hist(const int* __restrict__ Y,
                                              unsigned* __restrict__ counts, int m) {
    __shared__ unsigned h[K_CLASSES];
    if (threadIdx.x < K_CLASSES) h[threadIdx.x] = 0u;
    __syncthreads();
    int i = blockIdx.x * 256 + threadIdx.x;
    if (i < m) atomicAdd(&h[Y[i] & (K_CLASSES - 1)], 1u);
    __syncthreads();
    if (threadIdx.x < K_CLASSES && h[threadIdx.x])
        atomicAdd(&counts[threadIdx.x], h[threadIdx.x]);
}

// ---------------- kernel 3: column sums for Z_mean -------------------------
__global__ __launch_bounds__(256) void k_colsum(const float* __restrict__ Z,
                                                float* __restrict__ colsum, int m) {
    int col  = threadIdx.x & 127;
    int pair = threadIdx.x >> 7;          // 0 or 1
    int rowbase = blockIdx.x * 64;
    float acc = 0.f;
    for (int r = pair; r < 64; r += 2) {
        int row = rowbase + r;
        if (row < m) acc += Z[(size_t)row * D_FEAT + col];
    }
    unsafeAtomicAdd(&colsum[col], acc);
}

// ---------------- kernel 4: scan + Z_mean ----------------------------------
__global__ void k_scan(const unsigned* __restrict__ counts,
                       unsigned* __restrict__ offsets,
                       unsigned* __restrict__ cursor,
                       const float* __restrict__ colsum,
                       float* __restrict__ out_mean, int m) {
    if (threadIdx.x == 0) {
        unsigned acc = 0;
        for (int k = 0; k < K_CLASSES; ++k) {
            offsets[k] = acc;
            cursor[k]  = acc;
            acc += counts[k];
        }
    }
    if (threadIdx.x < D_FEAT)
        out_mean[threadIdx.x] = colsum[threadIdx.x] / (float)m;
}

// ---------------- kernel 5: counting-sort scatter (1 wave per row) ---------
__global__ __launch_bounds__(256) void k_scatter(const float* __restrict__ Z,
                                                 const int* __restrict__ Y,
                                                 unsigned* __restrict__ cursor,
                                                 float* __restrict__ Zs, int m) {
    int wave = threadIdx.x >> 5;
    int lane = threadIdx.x & 31;
    int row  = blockIdx.x * 8 + wave;
    if (row >= m) return;
    int k = Y[row] & (K_CLASSES - 1);
    int pos = 0;
    if (lane == 0) pos = (int)atomicAdd(&cursor[k], 1u);
    pos = __shfl(pos, 0, 32);
    const float4* src = (const float4*)(Z  + (size_t)row * D_FEAT);
    float4*       dst = (float4*)(Zs + (size_t)pos * D_FEAT);
    dst[lane] = src[lane];   // 32 lanes x float4 = 128 floats
}

// ---------------- kernel 6: grouped Gram via WMMA f32 16x16x4 --------------
// block = 256 threads (8 waves). grid = (K_CLASSES, NCHUNKY).
// Double-buffered LDS staging via GLOBAL_LOAD_ASYNC_TO_LDS_B128 (ASYNCcnt),
// chunk n+1 streams in while chunk n feeds the WMMA pipe.
#define CHUNK 60           // samples per LDS buffer (multiple of 4)
#define LST   136          // padded LDS row stride (floats) -> conflict-free
__device__ __forceinline__ void stage_chunk(const float* __restrict__ src,   // Zs + base*128
                                            unsigned cnt, unsigned row0,
                                            float* __restrict__ buf) {
    for (int e = threadIdx.x; e < CHUNK * 32; e += 256) {
        int row = e >> 5;
        int c4  = (e & 31) << 2;
        unsigned gr = row0 + (unsigned)row;
        float* dst = &buf[row * LST + c4];
        if (gr < cnt) {
#if HAVE_ASYNC_LDS
            __builtin_amdgcn_global_load_async_to_lds_b128(
                (AS1 v4i*)(unsigned long long)(size_t)(src + (size_t)gr * D_FEAT + c4),
                (AS3 v4i*)(unsigned)(size_t)dst, 0, 0);
#else
            *(float4*)dst = *(const float4*)(src + (size_t)gr * D_FEAT + c4);
#endif
        } else {
            *(float4*)dst = make_float4(0.f, 0.f, 0.f, 0.f);
        }
    }
}

__global__ __launch_bounds__(256) void k_gram(const float* __restrict__ Zs,
                                              const unsigned* __restrict__ offsets,
                                              const unsigned* __restrict__ counts,
                                              float* __restrict__ outZ) {
    __shared__ float sz[2][CHUNK * LST];   // 2 x 32640 B = 65280 B
    const int k         = blockIdx.x;
    const unsigned base = offsets[k];
    const unsigned cnt  = counts[k];
    const int wave = threadIdx.x >> 5;
    const int lane = threadIdx.x & 31;
    const int lr   = lane & 15;
    const int r    = wave * 16;
    const float* ZsBase = Zs + (size_t)base * D_FEAT;

    v8f acc[8];
#pragma unroll
    for (int t = 0; t < 8; ++t) acc[t] = (v8f){0.f,0.f,0.f,0.f,0.f,0.f,0.f,0.f};

    unsigned j = blockIdx.y;               // chunk index, strided by gridDim.y
    const unsigned jstride = gridDim.y;
    int cur = 0;
    if (j * (unsigned)CHUNK < cnt)
        stage_chunk(ZsBase, cnt, j * CHUNK, sz[0]);

    while (j * (unsigned)CHUNK < cnt) {
        wait_async0();                     // our buffer's async loads done
        __syncthreads();                   // all waves' loads visible; prev readers done
        unsigned jn = j + jstride;
        if (jn * (unsigned)CHUNK < cnt)    // prefetch next chunk into other buffer
            stage_chunk(ZsBase, cnt, jn * CHUNK, sz[cur ^ 1]);

        // K-loop: 4 samples per wmma. A[M][K]=Zs[s][r+M], B[K][N]=Zs[s][c+N]
        // f32 16x4 A layout: lanes 0-15 hold K=0,1 (vgpr0,1); lanes 16-31 hold K=2,3.
        const float* bufc = sz[cur];
        for (int s = 0; s < CHUNK; s += 4) {
            int ss = (lane < 16) ? s : s + 2;
            const float* p0 = &bufc[ss * LST];
            const float* p1 = &bufc[(ss + 1) * LST];
            v2f a; a.x = p0[r + lr]; a.y = p1[r + lr];
#pragma unroll
            for (int t = 0; t < 8; ++t) {
                int fc = t * 16 + lr;
                v2f b; b.x = p0[fc]; b.y = p1[fc];
                acc[t] = __builtin_amdgcn_wmma_f32_16x16x4_f32(
                    false, a, false, b, (short)0, acc[t], false, false);
            }
        }
        cur ^= 1;
        j = jn;
    }

    // split-K partial: atomically add C tiles into ZtPiZ[k]
    // C layout: lanes 0-15 -> M = vgpr j, N = lane; lanes 16-31 -> M = j+8, N = lane-16
    float* dstk = outZ + (size_t)k * D_FEAT * D_FEAT;
    const int mb = (lane < 16) ? 0 : 8;
#pragma unroll
    for (int t = 0; t < 8; ++t) {
        int gc = t * 16 + lr;
#pragma unroll
        for (int jj = 0; jj < 8; ++jj) {
            int gr = r + mb + jj;
            unsafeAtomicAdd(&dstk[gr * D_FEAT + gc], acc[t][jj]);
        }
    }
}

// ---------------- kernel 7: 33 x (128x128) Cholesky logdet -----------------
__global__ __launch_bounds__(128) void k_logdet(const float* __restrict__ ZtPiZ,
                                                const unsigned* __restrict__ counts,
                                                float* __restrict__ ld, int m) {
    __shared__ float A[D_FEAT * D_FEAT];   // 65536 B, stride 128
    const int b   = blockIdx.x;
    const int tid = threadIdx.x;

    if (b == 0) {
        // discriminative term: I + gam1*d/(m*eps) * sum_k ZtPiZ[k]
        const float cg = GAM1f * (float)D_FEAT / ((float)m * EPSf);
        for (int idx = tid; idx < D_FEAT * D_FEAT; idx += 128) {
            float s = 0.f;
            for (int k = 0; k < K_CLASSES; ++k)
                s += ZtPiZ[(size_t)k * D_FEAT * D_FEAT + idx];
            int row = idx >> 7, col = idx & 127;
            A[idx] = ((row == col) ? 1.f : 0.f) + cg * s;
        }
    } else {
        const int k = b - 1;
        const float nk = (float)counts[k] + 1e-8f;
        const float sc = (float)D_FEAT / (nk * EPSf);
        const float* src = ZtPiZ + (size_t)k * D_FEAT * D_FEAT;
        for (int idx = tid; idx < D_FEAT * D_FEAT; idx += 128) {
            int row = idx >> 7, col = idx & 127;
            A[idx] = ((row == col) ? 1.f : 0.f) + sc * src[idx];
        }
    }
    __syncthreads();

    // in-place right-looking Cholesky, one row per thread
    for (int kk = 0; kk < D_FEAT; ++kk) {
        if (tid == kk) A[kk * D_FEAT + kk] = sqrtf(A[kk * D_FEAT + kk]);
        __syncthreads();
        float akk = A[kk * D_FEAT + kk];
        if (tid > kk) A[tid * D_FEAT + kk] /= akk;
        __syncthreads();
        if (tid > kk) {
            float lik = A[tid * D_FEAT + kk];
            for (int j = kk + 1; j <= tid; ++j)
                A[tid * D_FEAT + j] -= lik * A[j * D_FEAT + kk];
        }
        __syncthreads();
    }

    // logdet = 2 * sum(log(diag)); reuse row 0 of A for the reduction
    float v = logf(A[tid * D_FEAT + tid]);
    __syncthreads();
    A[tid] = v;
    __syncthreads();
    for (int s2 = 64; s2 > 0; s2 >>= 1) {
        if (tid < s2) A[tid] += A[tid + s2];
        __syncthreads();
    }
    if (tid == 0) ld[b] = 2.f * A[0];
}

// ---------------- kernel 8: combine into scalar loss -----------------------
__global__ void k_final(const float* __restrict__ ld,
                        const unsigned* __restrict__ counts,
                        float* __restrict__ out, int m) {
    if (threadIdx.x == 0 && blockIdx.x == 0) {
        float discrim = ld[0] * 0.5f;
        float comp = 0.f;
        for (int k = 0; k < K_CLASSES; ++k)
            comp += ld[1 + k] * (((float)counts[k] + 1e-8f) / (float)m);
        comp *= 0.5f;
        out[0] = GAM2f * (-discrim) + comp;
    }
}

extern "C" void kernel_launch(void* const* d_in, const int* in_sizes, int n_in,
                              void* d_out, int out_size, void* d_ws, size_t ws_size,
                              hipStream_t stream) {
    const float* Z = (const float*)d_in[0];
    const int*   Y = (const int*)d_in[1];
    float* out = (float*)d_out;
    const int m = in_sizes[0] / D_FEAT;

    // workspace layout (first 512 floats = control block, zeroed each call)
    unsigned* counts  = (unsigned*)d_ws;          // [32]
    unsigned* offsets = counts + 32;              // [32]
    unsigned* cursor  = counts + 64;              // [32]
    float*    wsf     = (float*)d_ws;
    float*    colsum  = wsf + 96;                 // [128]
    float*    ld      = wsf + 224;                // [33]
    float*    Zs      = wsf + 512;                // [m*128] sorted copy (32 MB)

    // d_out layout: [0]=loss, [1..1+32*128*128)=ZtPiZ, then [128]=Z_mean
    float* outZ    = out + 1;
    float* outMean = out + 1 + K_CLASSES * D_FEAT * D_FEAT;

    k_zero   <<<1024, 256, 0, stream>>>(out, out_size, wsf);
    k_hist   <<<(m + 255) / 256, 256, 0, stream>>>(Y, counts, m);
    k_colsum <<<(m + 63) / 64,   256, 0, stream>>>(Z, colsum, m);
    k_scan   <<<1, 128, 0, stream>>>(counts, offsets, cursor, colsum, outMean, m);
    k_scatter<<<(m + 7) / 8,     256, 0, stream>>>(Z, Y, cursor, Zs, m);
    k_gram   <<<dim3(K_CLASSES, 16), 256, 0, stream>>>(Zs, offsets, counts, outZ);
    k_logdet <<<K_CLASSES + 1, 128, 0, stream>>>(outZ, counts, ld, m);
    k_final  <<<1, 32, 0, stream>>>(ld, counts, out, m);
}